// ResNetConv1D_10161892622550
// MI455X (gfx1250) — compile-verified
//
#include <hip/hip_runtime.h>

// ResNetConv1D on MI455X (gfx1250, wave32).
// Fully fused 12-block residual 2-tap conv stencil:
//   - one workgroup per row (4096 rows, L = 8192 f32)
//   - row DMA'd global->LDS via CDNA5 async-tensor path (ASYNCcnt)
//   - per-thread 32-element register tile; 2-float halo exchange per block
//   - exactly one global read + one global write per element (256 MB total)

#define BATCHN   4096
#define LROW     8192
#define NBLOCKS  12
#define NT       256
#define EPT      (LROW / NT)   // 32 elements per thread
#define V4PT     (EPT / 4)     // 8 float4 chunks per thread
#define ALPHA_F  0.9f

__device__ __forceinline__ float elu1(float t) {
    // elu(t) = t > 0 ? t : exp(t) - 1   (one v_exp_f32 per element)
    float e = __expf(t) - 1.0f;
    return t > 0.0f ? t : e;
}

__global__ void __launch_bounds__(NT) resconv12_kernel(
    const float* __restrict__ x,
    const float* __restrict__ w1, const float* __restrict__ b1,
    const float* __restrict__ w2, const float* __restrict__ b2,
    float* __restrict__ out)
{
    __shared__ __align__(16) float xs[LROW]; // 32 KB staging for async DMA
    __shared__ float hL[NT];                 // each thread's last element
    __shared__ float hR[NT];                 // each thread's first element

    const int t = threadIdx.x;
    const long long row = blockIdx.x;
    const float* __restrict__ xrow = x + row * (long long)LROW;
    float* __restrict__ orow = out + row * (long long)LROW;

    // ---- Stage 1: async DMA global row -> LDS (gfx1250 ASYNCcnt path) ----
    {
        const char* gbase = (const char*)xrow;
        #pragma unroll
        for (int j = 0; j < V4PT; ++j) {
            const int c = t + j * NT;  // linear 16B chunks: coalesced stream
            unsigned int lds = (unsigned int)(unsigned long long)(uintptr_t)(xs + 4 * c);
            unsigned long long ga = (unsigned long long)(uintptr_t)(gbase + 16ull * (unsigned long long)c);
            asm volatile("global_load_async_to_lds_b128 %0, %1, off"
                         :: "v"(lds), "v"(ga)
                         : "memory");
        }
        asm volatile("s_wait_asynccnt 0" ::: "memory");
    }
    __syncthreads();

    // ---- Stage 2: LDS -> registers (thread t owns x[32t .. 32t+31]) ----
    float xr[EPT];
    {
        const float4* xs4 = (const float4*)xs;
        #pragma unroll
        for (int k = 0; k < V4PT; ++k) {
            float4 v = xs4[t * V4PT + k];
            xr[4 * k + 0] = v.x;
            xr[4 * k + 1] = v.y;
            xr[4 * k + 2] = v.z;
            xr[4 * k + 3] = v.w;
        }
    }

    // ---- Stage 3: 12 residual conv blocks, register-resident ----
    for (int b = 0; b < NBLOCKS; ++b) {
        const float a0  = w1[2 * b + 0];
        const float a1  = w1[2 * b + 1];
        const float bb1 = b1[b];
        // fold ALPHA into second conv: x += ac0*y[i] + ac1*y[i+1] + ab2
        const float ac0 = ALPHA_F * w2[2 * b + 0];
        const float ac1 = ALPHA_F * w2[2 * b + 1];
        const float ab2 = ALPHA_F * b2[b];

        // publish radius-1 halo (2 floats per thread)
        hL[t] = xr[EPT - 1];
        hR[t] = xr[0];
        __syncthreads();
        const float xm1 = (t > 0)      ? hL[t - 1] : 0.0f; // x[base-1] (0-pad at i=0)
        const float xp1 = (t < NT - 1) ? hR[t + 1] : 0.0f; // x[base+EPT] (0-pad at i=L)

        // y[i] = elu(a0*x[i-1] + a1*x[i] + b1),  i = base .. base+EPT
        // x[i] += ac0*y[i] + ac1*y[i+1] + ab2
        float yprev = elu1(fmaf(a0, xm1, fmaf(a1, xr[0], bb1)));
        #pragma unroll
        for (int k = 0; k < EPT - 1; ++k) {
            const float ycur = elu1(fmaf(a0, xr[k], fmaf(a1, xr[k + 1], bb1)));
            xr[k] = fmaf(ac0, yprev, fmaf(ac1, ycur, xr[k] + ab2));
            yprev = ycur;
        }
        {
            const float ycur = elu1(fmaf(a0, xr[EPT - 1], fmaf(a1, xp1, bb1)));
            xr[EPT - 1] = fmaf(ac0, yprev, fmaf(ac1, ycur, xr[EPT - 1] + ab2));
        }
        __syncthreads(); // protect hL/hR before next block rewrites them
    }

    // ---- Stage 4: registers -> global (B128 stores) ----
    {
        float4* o4 = (float4*)orow;
        #pragma unroll
        for (int k = 0; k < V4PT; ++k) {
            float4 v;
            v.x = xr[4 * k + 0];
            v.y = xr[4 * k + 1];
            v.z = xr[4 * k + 2];
            v.w = xr[4 * k + 3];
            o4[t * V4PT + k] = v;
        }
    }
}

extern "C" void kernel_launch(void* const* d_in, const int* in_sizes, int n_in,
                              void* d_out, int out_size, void* d_ws, size_t ws_size,
                              hipStream_t stream) {
    const float* x  = (const float*)d_in[0];
    const float* w1 = (const float*)d_in[1];
    const float* b1 = (const float*)d_in[2];
    const float* w2 = (const float*)d_in[3];
    const float* b2 = (const float*)d_in[4];
    float* out = (float*)d_out;

    resconv12_kernel<<<dim3(BATCHN), dim3(NT), 0, stream>>>(x, w1, b1, w2, b2, out);
}